// Infomax_14955076125207
// MI455X (gfx1250) — compile-verified
//
#include <hip/hip_runtime.h>
#include <hip/hip_bf16.h>

#define N_NODES 100000
#define N_EDGES 1600000
#define IN_DIM  128
#define HID     64
#define NTILES  (N_NODES / 16)   // 6250, exact

typedef float v2f __attribute__((ext_vector_type(2)));
typedef float v8f __attribute__((ext_vector_type(8)));

// ---------------------------------------------------------------------------
// Utility fill kernel
// ---------------------------------------------------------------------------
__global__ void dgi_fill(float* __restrict__ p, float v, size_t n) {
    size_t i = (size_t)blockIdx.x * blockDim.x + threadIdx.x;
    if (i < n) p[i] = v;
}

// ---------------------------------------------------------------------------
// Degree count: deg[dst]++ per edge (deg pre-filled with 1.0 for self loop)
// ---------------------------------------------------------------------------
__global__ void dgi_count_deg(const int* __restrict__ dst, float* __restrict__ deg) {
    int e = blockIdx.x * 256 + threadIdx.x;
    if (e < N_EDGES) unsafeAtomicAdd(&deg[dst[e]], 1.0f);
}

__global__ void dgi_rsqrt_ip(float* __restrict__ d) {
    int i = blockIdx.x * 256 + threadIdx.x;
    if (i < N_NODES) d[i] = rsqrtf(d[i]);
}

// ---------------------------------------------------------------------------
// h = gather(x, rowmap) @ W   using V_WMMA_F32_16X16X4_F32
// One wave -> 16 rows x 64 cols. blockDim = 256 (8 waves).
// A frag (16x4 f32): lanes 0-15 hold M=lane, v={K=k,k+1}; lanes 16-31 K={k+2,k+3}
// B frag (4x16 f32): mirrored with N=lane&15
// C/D (16x16 f32): VGPR r -> M = 8*(lane>>4)+r, N = lane&15 (per ISA 7.12.2)
//
// W is staged in LDS pre-paired as float2 {W[2p][c], W[2p+1][c]} so every B
// fragment is ONE aligned ds_load_b64 straight into an even VGPR pair
// (WMMA SRC alignment), avoiding v_dual_mov repacking in the hot loop.
// ---------------------------------------------------------------------------
__global__ void __launch_bounds__(256)
dgi_gemm_wmma(const float* __restrict__ x, const float* __restrict__ W,
              const int* __restrict__ rowmap, float* __restrict__ h) {
    __shared__ v2f WsP[(IN_DIM / 2) * HID];         // 64 pair-rows x 64 cols = 32 KB
    for (int i = threadIdx.x; i < (IN_DIM / 2) * HID; i += 256) {
        const int p = i >> 6;                        // pair row (covers K=2p,2p+1)
        const int c = i & 63;
        v2f w;
        w.x = W[(2 * p)     * HID + c];
        w.y = W[(2 * p + 1) * HID + c];
        WsP[i] = w;
    }
    __syncthreads();

    const int wave    = threadIdx.x >> 5;
    const int lane    = threadIdx.x & 31;
    const int tile    = blockIdx.x * 8 + wave;
    if (tile >= NTILES) return;

    const int halfsel = lane >> 4;                  // 0 or 1
    const int lrow    = lane & 15;
    const int row     = tile * 16 + lrow;
    const int srow    = rowmap ? rowmap[row] : row;
    const float* xr   = x + (size_t)srow * IN_DIM;

    v8f acc0 = {}, acc1 = {}, acc2 = {}, acc3 = {};

    for (int k = 0; k < IN_DIM; k += 4) {
        const int kk = k + (halfsel << 1);
        v2f a = *(const v2f*)(xr + kk);             // 8B aligned (kk even)
        const v2f* wp = &WsP[((k >> 1) + halfsel) * HID + lrow];
        v2f b0 = wp[0];                             // single ds_load_b64 each
        v2f b1 = wp[16];
        v2f b2 = wp[32];
        v2f b3 = wp[48];
        acc0 = __builtin_amdgcn_wmma_f32_16x16x4_f32(false, a, false, b0, (short)0, acc0, false, false);
        acc1 = __builtin_amdgcn_wmma_f32_16x16x4_f32(false, a, false, b1, (short)0, acc1, false, false);
        acc2 = __builtin_amdgcn_wmma_f32_16x16x4_f32(false, a, false, b2, (short)0, acc2, false, false);
        acc3 = __builtin_amdgcn_wmma_f32_16x16x4_f32(false, a, false, b3, (short)0, acc3, false, false);
    }

    const int rbase = tile * 16 + halfsel * 8;
#pragma unroll
    for (int r = 0; r < 8; ++r) {
        float* hr = h + (size_t)(rbase + r) * HID + lrow;
        hr[0]  = acc0[r];
        hr[16] = acc1[r];
        hr[32] = acc2[r];
        hr[48] = acc3[r];
    }
}

// ---------------------------------------------------------------------------
// Edge scatter: out[dst] += h[src] * dinv[src]*dinv[dst]
// 4 edges per block of 256; edge index loads are wave-uniform -> scalar loads.
// ---------------------------------------------------------------------------
__global__ void __launch_bounds__(256)
dgi_scatter(const int* __restrict__ src, const int* __restrict__ dst,
            const float* __restrict__ dinv, const float* __restrict__ h,
            float* __restrict__ out) {
    const int e = blockIdx.x * 4 + (threadIdx.x >> 6);
    const int c = threadIdx.x & 63;
    if (e >= N_EDGES) return;
    const int s = src[e];
    const int d = dst[e];
    const float norm = dinv[s] * dinv[d];
    unsafeAtomicAdd(&out[(size_t)d * HID + c], h[(size_t)s * HID + c] * norm);
}

// ---------------------------------------------------------------------------
// Finalize: agg = prelu(agg + h*dinv^2 + bias); optional column-sum for summary
// 4 nodes per 256-thread block.
// ---------------------------------------------------------------------------
__global__ void __launch_bounds__(256)
dgi_finalize(float* __restrict__ agg, const float* __restrict__ h,
             const float* __restrict__ dinv, const float* __restrict__ bias,
             const float* __restrict__ pa, float* __restrict__ colsum, int doSum) {
    __shared__ float sh[HID];
    const int c  = threadIdx.x & 63;
    const int li = threadIdx.x >> 6;
    const int i  = blockIdx.x * 4 + li;
    if (threadIdx.x < HID) sh[threadIdx.x] = 0.0f;
    __syncthreads();
    float p = 0.0f;
    if (i < N_NODES) {
        const float di  = dinv[i];
        const size_t ix = (size_t)i * HID + c;
        const float v   = agg[ix] + h[ix] * di * di + bias[c];
        p = (v > 0.0f) ? v : v * pa[c];
        agg[ix] = p;
    }
    if (doSum) {
        atomicAdd(&sh[c], p);                 // ds_add_f32
        __syncthreads();
        if (threadIdx.x < HID) unsafeAtomicAdd(&colsum[threadIdx.x], sh[threadIdx.x]);
    }
}

// ---------------------------------------------------------------------------
// s = W_disc @ sigmoid(colsum / N).  One block of 64 threads.
// ---------------------------------------------------------------------------
__global__ void dgi_make_s(const float* __restrict__ colsum,
                           const float* __restrict__ Wd, float* __restrict__ s) {
    __shared__ float sm[HID];
    const int t = threadIdx.x;
    const float m = colsum[t] * (1.0f / N_NODES);
    sm[t] = 1.0f / (1.0f + expf(-m));
    __syncthreads();
    float a = 0.0f;
#pragma unroll
    for (int c = 0; c < HID; ++c) a += Wd[t * HID + c] * sm[c];
    s[t] = a;
}

// ---------------------------------------------------------------------------
// Accumulate sum_i softplus(sign * (mat[i] . s)) into acc[0]
// ---------------------------------------------------------------------------
__global__ void __launch_bounds__(256)
dgi_loss(const float* __restrict__ mat, const float* __restrict__ s,
         float* __restrict__ acc, float sign) {
    __shared__ float ssh[HID];
    __shared__ float red[256];
    const int t = threadIdx.x;
    if (t < HID) ssh[t] = s[t];
    __syncthreads();
    const int i = blockIdx.x * 256 + t;
    float val = 0.0f;
    if (i < N_NODES) {
        const float* p = mat + (size_t)i * HID;
        float dp = 0.0f;
#pragma unroll
        for (int c = 0; c < HID; ++c) dp += p[c] * ssh[c];
        const float xl = sign * dp;
        val = fmaxf(xl, 0.0f) + log1pf(expf(-fabsf(xl)));   // stable softplus
    }
    red[t] = val;
    __syncthreads();
    for (int off = 128; off > 0; off >>= 1) {
        if (t < off) red[t] += red[t + off];
        __syncthreads();
    }
    if (t == 0) unsafeAtomicAdd(acc, red[0]);
}

__global__ void dgi_finish(const float* __restrict__ acc, float* __restrict__ out) {
    out[0] = acc[0] * (1.0f / N_NODES);
}

// ---------------------------------------------------------------------------
// Host launcher
// ---------------------------------------------------------------------------
extern "C" void kernel_launch(void* const* d_in, const int* in_sizes, int n_in,
                              void* d_out, int out_size, void* d_ws, size_t ws_size,
                              hipStream_t stream) {
    const float* x      = (const float*)d_in[0];   // [N, 128]
    const float* W_gcn  = (const float*)d_in[1];   // [128, 64]
    const float* b_gcn  = (const float*)d_in[2];   // [64]
    const float* prelu  = (const float*)d_in[3];   // [64]
    const float* W_disc = (const float*)d_in[4];   // [64, 64]
    const int*   ei     = (const int*)d_in[5];     // [2, E] (int32 on device)
    const int*   perm   = (const int*)d_in[6];     // [N]
    const int* src = ei;
    const int* dst = ei + N_EDGES;

    const size_t NH = (size_t)N_NODES * HID;       // 6.4M floats
    float* ws    = (float*)d_ws;
    float* hbuf  = ws;                 // [NH]  h = x@W (reused pos/neg)
    float* agg   = ws + NH;            // [NH]  aggregation -> prelu output
    float* csum  = ws + 2 * NH;        // [64]  column sums for summary
    float* accp  = csum + HID;         // [16]  loss accumulator
    float* svec  = accp + 16;          // [64]  s = W_disc @ summary
    float* dinv  = svec + HID;         // [N]   deg -> deg^{-1/2} in place

    const int nodeBlocks = (N_NODES + 255) / 256;          // 391
    const int edgeBlocks = (N_EDGES + 255) / 256;          // 6250
    const int nhBlocks   = (int)((NH + 255) / 256);        // 25000
    const int gemmBlocks = (NTILES + 7) / 8;               // 782
    const int scatBlocks = N_EDGES / 4;                    // 400000
    const int finBlocks  = N_NODES / 4;                    // 25000

    // ---- degree / normalization (shared by pos & neg views) ----
    dgi_fill<<<nodeBlocks, 256, 0, stream>>>(dinv, 1.0f, (size_t)N_NODES);
    dgi_count_deg<<<edgeBlocks, 256, 0, stream>>>(dst, dinv);
    dgi_rsqrt_ip<<<nodeBlocks, 256, 0, stream>>>(dinv);
    dgi_fill<<<1, 128, 0, stream>>>(csum, 0.0f, (size_t)(HID + 16));  // csum + acc

    // ---- positive view ----
    dgi_fill<<<nhBlocks, 256, 0, stream>>>(agg, 0.0f, NH);
    dgi_gemm_wmma<<<gemmBlocks, 256, 0, stream>>>(x, W_gcn, nullptr, hbuf);
    dgi_scatter<<<scatBlocks, 256, 0, stream>>>(src, dst, dinv, hbuf, agg);
    dgi_finalize<<<finBlocks, 256, 0, stream>>>(agg, hbuf, dinv, b_gcn, prelu, csum, 1);
    dgi_make_s<<<1, 64, 0, stream>>>(csum, W_disc, svec);
    dgi_loss<<<nodeBlocks, 256, 0, stream>>>(agg, svec, accp, -1.0f);

    // ---- negative (corrupted) view: same graph, permuted features ----
    dgi_fill<<<nhBlocks, 256, 0, stream>>>(agg, 0.0f, NH);
    dgi_gemm_wmma<<<gemmBlocks, 256, 0, stream>>>(x, W_gcn, perm, hbuf);
    dgi_scatter<<<scatBlocks, 256, 0, stream>>>(src, dst, dinv, hbuf, agg);
    dgi_finalize<<<finBlocks, 256, 0, stream>>>(agg, hbuf, dinv, b_gcn, prelu, csum, 0);
    dgi_loss<<<nodeBlocks, 256, 0, stream>>>(agg, svec, accp, +1.0f);

    dgi_finish<<<1, 1, 0, stream>>>(accp, (float*)d_out);
}